// Gating_72713796321589
// MI455X (gfx1250) — compile-verified
//
#include <hip/hip_runtime.h>

typedef __attribute__((ext_vector_type(2))) float v2f;
typedef __attribute__((ext_vector_type(8))) float v8f;

#define E_NUM   4
#define HID     512
#define T_TOTAL 65536
#define WPB     8      // waves per block
#define TPW     16     // tokens per wave (WMMA M)

__global__ __launch_bounds__(256) void moe_gating_wmma(
    const float* __restrict__ x,        // [T, HID]
    const float* __restrict__ gw,       // [E, HID]
    const float* __restrict__ gb,       // [E]
    float* __restrict__ sparse_out,     // [T, E]
    int*   __restrict__ idx_out,        // [T, 2]
    float* __restrict__ logit_out)      // [T, E]
{
    const int lane = threadIdx.x & 31;
    const int wave = threadIdx.x >> 5;
    const int sub  = lane & 15;         // M row (A) / N col (B,D) within half
    const int hi   = lane >> 4;         // half-wave select

    const int tokBase = (blockIdx.x * WPB + wave) * TPW;

    // A fragment: lane holds x[tokBase+sub][k + 2*hi .. +1]  (float2, contiguous)
    const float* xp = x + (long)(tokBase + sub) * HID + 2 * hi;
    // B fragment: expert column n = sub; columns 4..15 duplicate experts 0..3
    // (their D columns are never read), so no masking / divergence needed.
    const float* wp = gw + (long)(sub & 3) * HID + 2 * hi;

    v8f acc = {};   // 16x16 fp32 accumulator, 8 VGPRs

    #pragma unroll 8
    for (int k = 0; k < HID; k += 4) {
        v2f a = *(const v2f*)(xp + k);
        v2f b = *(const v2f*)(wp + k);
        acc = __builtin_amdgcn_wmma_f32_16x16x4_f32(
            /*neg_a=*/false, a, /*neg_b=*/false, b,
            /*c_mod=*/(short)0, acc, /*reuse_a=*/false, /*reuse_b=*/false);
    }

    const float bias = gb[sub & 3];

    #pragma unroll
    for (int r = 0; r < 8; ++r) {
        // D layout: VGPR r, lanes 0-15 -> token (tokBase+r), lanes 16-31 -> (tokBase+r+8);
        // N (expert) = lane & 15.
        float vr = acc[r] + bias;
        const int base = hi << 4;
        float v0 = __shfl(vr, base + 0, 32);
        float v1 = __shfl(vr, base + 1, 32);
        float v2 = __shfl(vr, base + 2, 32);
        float v3 = __shfl(vr, base + 3, 32);

        // top-1 (strict > => lowest index wins ties, matching jax.lax.top_k)
        float m1 = v0; int i1 = 0;
        if (v1 > m1) { m1 = v1; i1 = 1; }
        if (v2 > m1) { m1 = v2; i1 = 2; }
        if (v3 > m1) { m1 = v3; i1 = 3; }
        // top-2 among the rest
        float m2 = -__builtin_inff(); int i2 = 0;
        if (i1 != 0 && v0 > m2) { m2 = v0; i2 = 0; }
        if (i1 != 1 && v1 > m2) { m2 = v1; i2 = 1; }
        if (i1 != 2 && v2 > m2) { m2 = v2; i2 = 2; }
        if (i1 != 3 && v3 > m2) { m2 = v3; i2 = 3; }

        // softmax over {m1, m2} (all other experts are -inf -> 0)
        float e2  = __expf(m2 - m1);
        float inv = 1.0f / (1.0f + e2);
        float p1  = inv;
        float p2  = e2 * inv;

        const int t = tokBase + r + (hi << 3);

        if (sub == 0) {
            *(float4*)(logit_out + (long)t * 4) = make_float4(v0, v1, v2, v3);
        } else if (sub == 1) {
            float s0 = (i1 == 0) ? p1 : ((i2 == 0) ? p2 : 0.0f);
            float s1 = (i1 == 1) ? p1 : ((i2 == 1) ? p2 : 0.0f);
            float s2 = (i1 == 2) ? p1 : ((i2 == 2) ? p2 : 0.0f);
            float s3 = (i1 == 3) ? p1 : ((i2 == 3) ? p2 : 0.0f);
            *(float4*)(sparse_out + (long)t * 4) = make_float4(s0, s1, s2, s3);
        } else if (sub == 2) {
            *(int2*)(idx_out + (long)t * 2) = make_int2(i1, i2);
        }
    }
}

extern "C" void kernel_launch(void* const* d_in, const int* in_sizes, int n_in,
                              void* d_out, int out_size, void* d_ws, size_t ws_size,
                              hipStream_t stream) {
    const float* x  = (const float*)d_in[0];   // [8,8192,512] fp32
    const float* gw = (const float*)d_in[1];   // [4,512] fp32
    const float* gb = (const float*)d_in[2];   // [4] fp32

    // d_out = sparse_logits [T,4] fp32 | indices [T,2] int32 | gate_logit [T,4] fp32
    float* sparse_out = (float*)d_out;
    int*   idx_out    = (int*)((float*)d_out + (long)T_TOTAL * 4);
    float* logit_out  = (float*)d_out + (long)T_TOTAL * 4 + (long)T_TOTAL * 2;

    dim3 grid(T_TOTAL / (WPB * TPW));   // 512 blocks, exact cover
    dim3 block(WPB * 32);               // 8 wave32
    moe_gating_wmma<<<grid, block, 0, stream>>>(x, gw, gb, sparse_out, idx_out, logit_out);
}